// ChamferLoss_1580547973964
// MI455X (gfx1250) — compile-verified
//
#include <hip/hip_runtime.h>

// Chamfer loss on MI455X (gfx1250, wave32).
// zz cross-term via V_WMMA_F32_16X16X4_F32 (K=3 padded to 4, full fp32).
// A operand pre-scaled by -2, C operand carries x^2[m], so the WMMA output is
// already x^2[m] - 2*zz[m][n]; per lane we only add y^2[n] and min-reduce.

typedef __attribute__((ext_vector_type(2))) float v2f;
typedef __attribute__((ext_vector_type(8))) float v8f;

#define BATCH 16
#define NPTS  4096
#define NT    (NPTS / 16)   // 256 column tiles / row tiles
#define WAVES (BATCH * NT)  // 4096 waves per direction

// One wave handles (b, rowTile): 16 rows of X against all 4096 points of Y.
// Writes sum over its 16 rows of min_col pdist into partial[wave].
__global__ __launch_bounds__(256) void chamfer_min_rows(
    const float* __restrict__ X,   // (B, N, 3) "row" points
    const float* __restrict__ Y,   // (B, N, 3) "col" points
    float* __restrict__ partial) { // WAVES floats
  const int lane   = threadIdx.x & 31;
  const int wave   = blockIdx.x * (blockDim.x >> 5) + (threadIdx.x >> 5);
  const int b      = wave / NT;
  const int rowT   = wave % NT;
  const int half   = lane >> 4;    // 0 -> K={0,1}, 1 -> K={2,pad}
  const int l15    = lane & 15;

  // Load this lane's x point (row = rowT*16 + l15); lanes L and L+16 share it.
  const float* xp = X + ((size_t)b * NPTS + (size_t)rowT * 16 + l15) * 3;
  const float x0 = xp[0], x1 = xp[1], x2c = xp[2];
  const float myx2 = x0 * x0 + x1 * x1 + x2c * x2c;

  // A' = -2 * X tile. f32 16x4 A layout: lane&15 = M, VGPR v + 2*half = K.
  v2f a;
  a.x = -2.0f * (half ? x2c : x0);
  a.y = -2.0f * (half ? 0.0f : x1);

  // C matrix: C[m][n] = x^2[m]. C/D layout: element v <-> m = v + 8*half.
  v8f cmat;
#pragma unroll
  for (int v = 0; v < 8; ++v)
    cmat[v] = __shfl(myx2, v + 8 * half, 32);  // row r lives in lanes r, r+16

  v8f rmin;
#pragma unroll
  for (int v = 0; v < 8; ++v) rmin[v] = 3.402823466e38f;

  const float* yb = Y + (size_t)b * NPTS * 3;
  for (int jt = 0; jt < NT; ++jt) {
    const float* yp = yb + ((size_t)jt * 16 + l15) * 3;
    const float y0 = yp[0], y1 = yp[1], y2c = yp[2];
    const float y2sum = y0 * y0 + y1 * y1 + y2c * y2c;

    // Prefetch next column tile (global_prefetch_b8); clamp on last iter.
    const int jn = (jt + 1 < NT) ? (jt + 1) : jt;
    __builtin_prefetch(yb + ((size_t)jn * 16 + l15) * 3, 0, 3);

    // B operand 4x16: lane&15 = N, VGPR v + 2*half = K (same build as A).
    v2f bop;
    bop.x = half ? y2c : y0;
    bop.y = half ? 0.0f : y1;

    // D = A'(-2x) * B(y) + C(x^2)  ==  x^2[m] - 2*<x_m, y_n>
    v8f d = __builtin_amdgcn_wmma_f32_16x16x4_f32(
        /*neg_a=*/false, a, /*neg_b=*/false, bop,
        /*c_mod=*/(short)0, cmat, /*reuse_a=*/false, /*reuse_b=*/false);

#pragma unroll
    for (int v = 0; v < 8; ++v) {
      const float pd = d[v] + y2sum;  // full pdist(m, n)
      rmin[v] = fminf(rmin[v], pd);   // running min over column tiles
    }
  }

  // Min over the 16 columns: butterfly within each 16-lane half.
#pragma unroll
  for (int m = 1; m <= 8; m <<= 1) {
#pragma unroll
    for (int v = 0; v < 8; ++v)
      rmin[v] = fminf(rmin[v], __shfl_xor(rmin[v], m, 32));
  }

  // Sum row-mins: low half holds rows 0..7, high half rows 8..15.
  float s = 0.0f;
#pragma unroll
  for (int v = 0; v < 8; ++v) s += rmin[v];
  s += __shfl_xor(s, 16, 32);
  if (lane == 0) partial[wave] = s;
}

// Deterministic final reduction: sum both directions' partials, divide by B*N.
__global__ __launch_bounds__(256) void chamfer_reduce(
    const float* __restrict__ pA, const float* __restrict__ pB,
    float* __restrict__ out) {
  __shared__ float sbuf[256];
  float s = 0.0f;
  for (int i = threadIdx.x; i < WAVES; i += 256) s += pA[i] + pB[i];
  sbuf[threadIdx.x] = s;
  __syncthreads();
  for (int off = 128; off > 0; off >>= 1) {
    if ((int)threadIdx.x < off) sbuf[threadIdx.x] += sbuf[threadIdx.x + off];
    __syncthreads();
  }
  if (threadIdx.x == 0) out[0] = sbuf[0] * (1.0f / (float)(BATCH * NPTS));
}

extern "C" void kernel_launch(void* const* d_in, const int* in_sizes, int n_in,
                              void* d_out, int out_size, void* d_ws,
                              size_t ws_size, hipStream_t stream) {
  const float* x = (const float*)d_in[0];
  const float* y = (const float*)d_in[1];
  float* out = (float*)d_out;
  float* pA = (float*)d_ws;          // 4096 floats: mean(min over axis=2) dir
  float* pB = pA + WAVES;            // 4096 floats: mean(min over axis=1) dir

  const int blocks = (WAVES * 32) / 256;  // 512 blocks, 8 waves each
  chamfer_min_rows<<<blocks, 256, 0, stream>>>(x, y, pA);  // min_j for each i
  chamfer_min_rows<<<blocks, 256, 0, stream>>>(y, x, pB);  // min_i for each j
  chamfer_reduce<<<1, 256, 0, stream>>>(pA, pB, out);
}